// GAT_26774826123548
// MI455X (gfx1250) — compile-verified
//
#include <hip/hip_runtime.h>
#include <cmath>

// Problem constants (match reference)
#define NNODES 20000
#define NEDGES 320000
#define NEP    (NEDGES + NNODES)   // 340000 edges incl. self loops
#define NHEADS 8
#define CDIM   64
#define NHID   512                  // NHEADS * CDIM
#define NGRAPH 64
#define NCLS   10

typedef float v2f __attribute__((ext_vector_type(2)));
typedef float v8f __attribute__((ext_vector_type(8)));

// ---------------------------------------------------------------------------
// fp32 WMMA GEMM:  C[M x Nout] = A[M x K] @ B[K x Nout]
// Block = 8 waves, each wave a 16(M) x 64(N) strip; all waves share the same
// 64-column B panel staged in LDS, stored N-MAJOR with row stride 68 dwords:
//   Bs[n*68 + k] , k in [0,64), n in [0,64)
// so each WMMA B operand pair (k,k+1) is one 8B-aligned ds_load_b64, and the
// 32 lanes' two-bank spans (4n+k mod 64) are pairwise disjoint -> conflict-free.
// WMMA operand layouts per ISA 7.12.2 (32-bit A 16x4 / B 4x16 / C 16x16).
// M%16==0, Nout%64==0, K%64==0 hold for all three layers.
// ---------------------------------------------------------------------------
#define BK      64     // K-chunk staged in LDS
#define NSTRIDE 68     // n-major LDS row stride (dwords)

__global__ void __launch_bounds__(256)
gat_gemm_wmma(const float* __restrict__ A, const float* __restrict__ Bm,
              float* __restrict__ Cm, int M, int K, int Nout) {
  __shared__ float Bs[64 * NSTRIDE];           // 17.4 KB of 320 KB WGP LDS
  const int tid    = threadIdx.x;
  const int lane   = tid & 31;
  const int wave   = tid >> 5;
  const int mtile  = blockIdx.x * 8 + wave;
  const bool active = (mtile * 16 < M);
  const int mt     = active ? mtile : 0;       // tail waves: redundant work, no store
  const int n0     = blockIdx.y * 64;
  const int lane15 = lane & 15;
  const int khalf  = (lane >> 4) << 1;         // 0 (lanes 0-15) or 2 (16-31)
  const int row    = mt * 16 + lane15;

  v8f acc0 = {}, acc1 = {}, acc2 = {}, acc3 = {};
  const float* arow  = A + (size_t)row * K + khalf;
  const float* bbase = &Bs[lane15 * NSTRIDE + khalf];

  for (int kc = 0; kc < K; kc += BK) {
    __syncthreads();                           // previous chunk's reads done
    #pragma unroll
    for (int i = 0; i < (BK * 64) / 256; i++) {  // 16 elems/thread
      int idx = i * 256 + tid;                 // 0..4095
      int kk  = idx >> 6;                      // 0..63 (k)
      int nn  = idx & 63;                      // 0..63 (n) -> coalesced global read
      Bs[nn * NSTRIDE + kk] = Bm[(size_t)(kc + kk) * Nout + n0 + nn];
    }
    __syncthreads();

    const float* ap = arow + kc;
    #pragma unroll 4
    for (int k4 = 0; k4 < BK; k4 += 4) {
      __builtin_prefetch(ap + BK, 0, 3);       // prefetch next A chunk toward GL2
      v2f a; a.x = ap[0]; a.y = ap[1];
      ap += 4;
      v2f b0 = *(const v2f*)(bbase + k4);                  // n = lane15
      v2f b1 = *(const v2f*)(bbase + k4 + 16 * NSTRIDE);   // n = lane15+16
      v2f b2 = *(const v2f*)(bbase + k4 + 32 * NSTRIDE);   // n = lane15+32
      v2f b3 = *(const v2f*)(bbase + k4 + 48 * NSTRIDE);   // n = lane15+48
      acc0 = __builtin_amdgcn_wmma_f32_16x16x4_f32(false, a, false, b0, (short)0, acc0, false, false);
      acc1 = __builtin_amdgcn_wmma_f32_16x16x4_f32(false, a, false, b1, (short)0, acc1, false, false);
      acc2 = __builtin_amdgcn_wmma_f32_16x16x4_f32(false, a, false, b2, (short)0, acc2, false, false);
      acc3 = __builtin_amdgcn_wmma_f32_16x16x4_f32(false, a, false, b3, (short)0, acc3, false, false);
    }
  }

  if (active) {
    const int rbase = mt * 16 + (lane >> 4) * 8;
    float* cp = Cm + (size_t)rbase * Nout + n0 + lane15;
    #pragma unroll
    for (int r = 0; r < 8; r++) {
      cp[(size_t)r * Nout +  0] = acc0[r];
      cp[(size_t)r * Nout + 16] = acc1[r];
      cp[(size_t)r * Nout + 32] = acc2[r];
      cp[(size_t)r * Nout + 48] = acc3[r];
    }
  }
}

// ---------------------------------------------------------------------------
// Float atomic max via sign-split trick (order-independent -> deterministic).
// ---------------------------------------------------------------------------
__device__ __forceinline__ void atomicMaxF(float* addr, float val) {
  val += 0.0f;                                  // canonicalize -0 -> +0
  if (val >= 0.0f)
    atomicMax((int*)addr, __float_as_int(val));
  else
    atomicMin((unsigned int*)addr, __float_as_uint(val));
}

// Per-layer init: m = -FLT_MAX, s = 0, acc = 0
__global__ void gat_init(float* __restrict__ m, float* __restrict__ s,
                         float* __restrict__ acc) {
  int i = blockIdx.x * blockDim.x + threadIdx.x;
  if (i < NNODES * NHEADS) { m[i] = -3.402823466e38f; s[i] = 0.0f; }
  if (i < NNODES * NHID)   acc[i] = 0.0f;
}

// Attention coefficients: als[n,h] = <xt[n,h,:], a_src[h,:]>, same for ald.
__global__ void gat_att(const float* __restrict__ xt,
                        const float* __restrict__ a_src,
                        const float* __restrict__ a_dst,
                        float* __restrict__ als, float* __restrict__ ald) {
  int idx = blockIdx.x * blockDim.x + threadIdx.x;    // n*NHEADS + h
  if (idx >= NNODES * NHEADS) return;
  int n = idx >> 3, h = idx & 7;
  const float* xp = xt + (size_t)n * NHID + h * CDIM;
  const float* as = a_src + h * CDIM;
  const float* ad = a_dst + h * CDIM;
  float s1 = 0.f, s2 = 0.f;
  #pragma unroll 8
  for (int c = 0; c < CDIM; c++) { float v = xp[c]; s1 += v * as[c]; s2 += v * ad[c]; }
  als[idx] = s1; ald[idx] = s2;
}

// Pass A: e = leaky_relu(als[src]+ald[dst], 0.2); store e; segment max by dst.
__global__ void gat_edge_max(const int* __restrict__ ei,
                             const float* __restrict__ als,
                             const float* __restrict__ ald,
                             float* __restrict__ ebuf, float* __restrict__ m) {
  int e = blockIdx.x * blockDim.x + threadIdx.x;
  if (e >= NEP) return;
  int src = (e < NEDGES) ? ei[e]          : (e - NEDGES);
  int dst = (e < NEDGES) ? ei[NEDGES + e] : (e - NEDGES);
  #pragma unroll
  for (int h = 0; h < NHEADS; h++) {
    float v = als[src * NHEADS + h] + ald[dst * NHEADS + h];
    v = (v > 0.f) ? v : 0.2f * v;
    ebuf[(size_t)e * NHEADS + h] = v;
    atomicMaxF(&m[dst * NHEADS + h], v);
  }
}

// Pass B: p = exp(e - m[dst]); store p; segment sum by dst.
__global__ void gat_edge_exp(const int* __restrict__ ei,
                             const float* __restrict__ m,
                             float* __restrict__ ebuf, float* __restrict__ s) {
  int e = blockIdx.x * blockDim.x + threadIdx.x;
  if (e >= NEP) return;
  int dst = (e < NEDGES) ? ei[NEDGES + e] : (e - NEDGES);
  #pragma unroll
  for (int h = 0; h < NHEADS; h++) {
    float p = expf(ebuf[(size_t)e * NHEADS + h] - m[dst * NHEADS + h]);
    ebuf[(size_t)e * NHEADS + h] = p;
    atomicAdd(&s[dst * NHEADS + h], p);
  }
}

// Pass C: acc[dst,h,c] += xt[src,h,c] * p/(s[dst,h]+1e-16).
// One thread per (edge, 4-channel chunk): 128 chunks per edge.
__global__ void gat_edge_agg(const int* __restrict__ ei,
                             const float* __restrict__ xt,
                             const float* __restrict__ ebuf,
                             const float* __restrict__ s,
                             float* __restrict__ acc) {
  int idx = blockIdx.x * blockDim.x + threadIdx.x;
  int t = idx & 127;             // chunk -> channels [t*4, t*4+4)
  int e = idx >> 7;
  if (e >= NEP) return;
  int src = (e < NEDGES) ? ei[e]          : (e - NEDGES);
  int dst = (e < NEDGES) ? ei[NEDGES + e] : (e - NEDGES);
  int h = t >> 4;                // head = (t*4)/64
  float alpha = ebuf[(size_t)e * NHEADS + h] / (s[dst * NHEADS + h] + 1e-16f);
  const float4 xv = *(const float4*)(xt + (size_t)src * NHID + t * 4);
  float* op = acc + (size_t)dst * NHID + t * 4;
  atomicAdd(op + 0, xv.x * alpha);
  atomicAdd(op + 1, xv.y * alpha);
  atomicAdd(op + 2, xv.z * alpha);
  atomicAdd(op + 3, xv.w * alpha);
}

// h = elu(acc + bias)
__global__ void gat_bias_elu(const float* __restrict__ acc,
                             const float* __restrict__ bias,
                             float* __restrict__ out) {
  int i = blockIdx.x * blockDim.x + threadIdx.x;
  if (i >= NNODES * NHID) return;
  float v = acc[i] + bias[i & (NHID - 1)];
  out[i] = (v > 0.f) ? v : (expf(v) - 1.0f);
}

__global__ void pool_init(float* __restrict__ hG, float* __restrict__ cnt) {
  int i = blockIdx.x * blockDim.x + threadIdx.x;
  if (i < NGRAPH * NHID) hG[i] = 0.f;
  if (i < NGRAPH)        cnt[i] = 0.f;
}

__global__ void pool_sum(const float* __restrict__ h, const int* __restrict__ batch,
                         float* __restrict__ hG, float* __restrict__ cnt) {
  int i = blockIdx.x * blockDim.x + threadIdx.x;
  if (i >= NNODES * NHID) return;
  int n = i >> 9, j = i & (NHID - 1);
  int g = batch[n];
  atomicAdd(&hG[g * NHID + j], h[i]);
  if (j == 0) atomicAdd(&cnt[g], 1.0f);
}

__global__ void pool_div(float* __restrict__ hG, const float* __restrict__ cnt) {
  int i = blockIdx.x * blockDim.x + threadIdx.x;
  if (i >= NGRAPH * NHID) return;
  float c = cnt[i >> 9];
  hG[i] /= (c < 1.f ? 1.f : c);
}

// logits = hG @ Wl + bl; softmax. One wave per graph (wave32).
__global__ void classifier(const float* __restrict__ hG, const float* __restrict__ Wl,
                           const float* __restrict__ bl, float* __restrict__ probs) {
  int g = blockIdx.x;
  int lane = threadIdx.x;
  float logit = -3.402823466e38f;
  if (lane < NCLS) {
    float a = bl[lane];
    const float* hp = hG + (size_t)g * NHID;
    for (int k = 0; k < NHID; k++) a += hp[k] * Wl[k * NCLS + lane];
    logit = a;
  }
  float mx = logit;
  for (int off = 16; off; off >>= 1) mx = fmaxf(mx, __shfl_xor(mx, off, 32));
  float ex = (lane < NCLS) ? expf(logit - mx) : 0.f;
  float sm = ex;
  for (int off = 16; off; off >>= 1) sm += __shfl_xor(sm, off, 32);
  if (lane < NCLS) probs[g * NCLS + lane] = ex / sm;
}

// ---------------------------------------------------------------------------
extern "C" void kernel_launch(void* const* d_in, const int* in_sizes, int n_in,
                              void* d_out, int out_size, void* d_ws, size_t ws_size,
                              hipStream_t stream) {
  const float* x     = (const float*)d_in[0];
  const int*   ei    = (const int*)d_in[1];
  const int*   batch = (const int*)d_in[2];
  const float* W[3]    = {(const float*)d_in[3],  (const float*)d_in[7],  (const float*)d_in[11]};
  const float* asrc[3] = {(const float*)d_in[4],  (const float*)d_in[8],  (const float*)d_in[12]};
  const float* adst[3] = {(const float*)d_in[5],  (const float*)d_in[9],  (const float*)d_in[13]};
  const float* bias[3] = {(const float*)d_in[6],  (const float*)d_in[10], (const float*)d_in[14]};
  const float* Wl = (const float*)d_in[15];
  const float* bl = (const float*)d_in[16];

  float* hG    = (float*)d_out;                 // [NGRAPH, NHID]
  float* probs = hG + (size_t)NGRAPH * NHID;    // [NGRAPH, NCLS]

  // Workspace carve (floats)
  float* xt   = (float*)d_ws;                       // N*HID  transformed feats
  float* acc  = xt   + (size_t)NNODES * NHID;       // N*HID  aggregation
  float* hbuf = acc  + (size_t)NNODES * NHID;       // N*HID  layer activations
  float* als  = hbuf + (size_t)NNODES * NHID;       // N*H
  float* ald  = als  + (size_t)NNODES * NHEADS;     // N*H
  float* m    = ald  + (size_t)NNODES * NHEADS;     // N*H segment max
  float* sbuf = m    + (size_t)NNODES * NHEADS;     // N*H segment sum
  float* ebuf = sbuf + (size_t)NNODES * NHEADS;     // EP*H edge scores/probs
  float* cnt  = ebuf + (size_t)NEP * NHEADS;        // NGRAPH

  const dim3 blk(256);
  const int gridNH   = (NNODES * NHID) / 256;       // 40000
  const int gridAtt  = (NNODES * NHEADS) / 256;     // 625
  const int gridEdge = (NEP + 255) / 256;           // 1329
  const int gridAgg  = (NEP * 128) / 256;           // 170000
  const dim3 gemmGrid((1250 + 7) / 8, NHID / 64);   // (157, 8)

  for (int layer = 0; layer < 3; layer++) {
    const float* Xin = (layer == 0) ? x : hbuf;
    const int K = (layer == 0) ? 256 : NHID;
    gat_init<<<gridNH, blk, 0, stream>>>(m, sbuf, acc);
    gat_gemm_wmma<<<gemmGrid, blk, 0, stream>>>(Xin, W[layer], xt, NNODES, K, NHID);
    gat_att<<<gridAtt, blk, 0, stream>>>(xt, asrc[layer], adst[layer], als, ald);
    gat_edge_max<<<gridEdge, blk, 0, stream>>>(ei, als, ald, ebuf, m);
    gat_edge_exp<<<gridEdge, blk, 0, stream>>>(ei, m, ebuf, sbuf);
    gat_edge_agg<<<gridAgg, blk, 0, stream>>>(ei, xt, ebuf, sbuf, acc);
    gat_bias_elu<<<gridNH, blk, 0, stream>>>(acc, bias[layer], hbuf);
  }

  pool_init<<<(NGRAPH * NHID) / 256, blk, 0, stream>>>(hG, cnt);
  pool_sum<<<gridNH, blk, 0, stream>>>(hbuf, batch, hG, cnt);
  pool_div<<<(NGRAPH * NHID) / 256, blk, 0, stream>>>(hG, cnt);
  classifier<<<NGRAPH, dim3(32), 0, stream>>>(hG, Wl, bl, probs);
}